// FastGCN_84138409329232
// MI455X (gfx1250) — compile-verified
//
#include <hip/hip_runtime.h>
#include <hip/hip_bf16.h>

typedef __attribute__((ext_vector_type(2))) float v2f;
typedef __attribute__((ext_vector_type(8))) float v8f;

#define IN_F  128
#define HID_F 128
#define OUT_F 40

// ---------------- degree / normalization ----------------

__global__ void k_fill1(float* __restrict__ deg, int n) {
    int i = blockIdx.x * blockDim.x + threadIdx.x;
    if (i < n) deg[i] = 1.0f;                       // self-loop contributes 1
}

__global__ void k_degcount(const int* __restrict__ dst, float* __restrict__ deg, int e) {
    int i = blockIdx.x * blockDim.x + threadIdx.x;
    if (i < e) unsafeAtomicAdd(&deg[dst[i]], 1.0f);
}

__global__ void k_rsqrt(float* __restrict__ dis, int n) {
    int i = blockIdx.x * blockDim.x + threadIdx.x;
    if (i < n) dis[i] = rsqrtf(dis[i]);             // deg >= 1 always
}

// ---------------- GEMM1: H1[N,128] = x[N,128] @ W1[128,128] (f32 WMMA) ----------------

__global__ __launch_bounds__(256) void k_gemm1(const float* __restrict__ x,
                                               const float* __restrict__ W,
                                               float* __restrict__ H, int n) {
    const int lane = threadIdx.x & 31;
    const int wave = threadIdx.x >> 5;          // 0..7 -> column tile
    const int m0   = blockIdx.x * 16;
    const int n0   = wave * 16;
    const int row  = lane & 15;
    const int kh   = (lane >> 4) << 1;          // 0 or 2 (K sub-offset)

    int mr = m0 + row; if (mr >= n) mr = n - 1; // clamp loads, keep EXEC uniform
    const float* xrow = x + (size_t)mr * IN_F;

    v8f c = {0.f,0.f,0.f,0.f,0.f,0.f,0.f,0.f};
    #pragma unroll
    for (int k = 0; k < IN_F; k += 4) {
        v2f a, b;
        a.x = xrow[k + kh];
        a.y = xrow[k + kh + 1];
        b.x = W[(size_t)(k + kh)     * HID_F + n0 + row];
        b.y = W[(size_t)(k + kh + 1) * HID_F + n0 + row];
        c = __builtin_amdgcn_wmma_f32_16x16x4_f32(false, a, false, b,
                                                  (short)0, c, false, false);
    }
    const int mh  = (lane >> 4) * 8;
    const int col = n0 + row;
    #pragma unroll
    for (int r = 0; r < 8; ++r) {
        int m = m0 + r + mh;
        if (m < n) H[(size_t)m * HID_F + col] = c[r];
    }
}

// ---------------- layer1 self-loop + bias init:  A1 = dis^2 * H1 + b1 ----------------

__global__ void k_init1(const float* __restrict__ H, const float* __restrict__ dis,
                        const float* __restrict__ b, float* __restrict__ A, int n) {
    int idx = blockIdx.x * blockDim.x + threadIdx.x;
    if (idx < n * HID_F) {
        int i = idx >> 7, f = idx & 127;
        float s = dis[i];
        A[idx] = H[idx] * (s * s) + b[f];
    }
}

// ---------------- layer1 edge scatter: A1[dst] += w * H1[src]  (wave per edge) ----------------

__global__ void k_scatter1(const int* __restrict__ src, const int* __restrict__ dst,
                           const float* __restrict__ dis, const float* __restrict__ H,
                           float* __restrict__ A, int e) {
    int t = blockIdx.x * blockDim.x + threadIdx.x;   // E*32 work items
    if (t >= e * 32) return;
    int eg = t >> 5;            // all 32 lanes of a wave share one edge
    int c  = t & 31;            // 4-float chunk within the 128-wide row
    int s  = src[eg], d = dst[eg];
    float w = dis[s] * dis[d];
    const float4 v = ((const float4*)(H + (size_t)s * HID_F))[c];
    float* out = A + (size_t)d * HID_F + c * 4;
    unsafeAtomicAdd(out + 0, v.x * w);
    unsafeAtomicAdd(out + 1, v.y * w);
    unsafeAtomicAdd(out + 2, v.z * w);
    unsafeAtomicAdd(out + 3, v.w * w);
}

// ---------------- GEMM2: H2[N,40] = relu(A1)[N,128] @ W2[128,40] (f32 WMMA) ----------------

__global__ __launch_bounds__(96) void k_gemm2(const float* __restrict__ A,
                                              const float* __restrict__ W,
                                              float* __restrict__ H, int n) {
    const int lane = threadIdx.x & 31;
    const int wave = threadIdx.x >> 5;          // 0..2 -> column tiles 0,16,32
    const int m0   = blockIdx.x * 16;
    const int n0   = wave * 16;
    const int row  = lane & 15;
    const int kh   = (lane >> 4) << 1;

    int mr = m0 + row; if (mr >= n) mr = n - 1;
    const float* arow = A + (size_t)mr * HID_F;
    const int col = n0 + row;
    const bool colok = (col < OUT_F);

    v8f c = {0.f,0.f,0.f,0.f,0.f,0.f,0.f,0.f};
    #pragma unroll
    for (int k = 0; k < HID_F; k += 4) {
        v2f a, b;
        a.x = fmaxf(arow[k + kh],     0.0f);    // relu folded into A load
        a.y = fmaxf(arow[k + kh + 1], 0.0f);
        b.x = colok ? W[(size_t)(k + kh)     * OUT_F + col] : 0.0f;
        b.y = colok ? W[(size_t)(k + kh + 1) * OUT_F + col] : 0.0f;
        c = __builtin_amdgcn_wmma_f32_16x16x4_f32(false, a, false, b,
                                                  (short)0, c, false, false);
    }
    const int mh = (lane >> 4) * 8;
    #pragma unroll
    for (int r = 0; r < 8; ++r) {
        int m = m0 + r + mh;
        if (m < n && colok) H[(size_t)m * OUT_F + col] = c[r];
    }
}

// ---------------- layer2 self-loop + bias init:  out = dis^2 * H2 + b2 ----------------

__global__ void k_init2(const float* __restrict__ H, const float* __restrict__ dis,
                        const float* __restrict__ b, float* __restrict__ O, int n) {
    int idx = blockIdx.x * blockDim.x + threadIdx.x;
    if (idx < n * OUT_F) {
        int i = idx / OUT_F, f = idx % OUT_F;
        float s = dis[i];
        O[idx] = H[idx] * (s * s) + b[f];
    }
}

// ---------------- layer2 edge scatter: out[dst] += w * H2[src] ----------------

__global__ void k_scatter2(const int* __restrict__ src, const int* __restrict__ dst,
                           const float* __restrict__ dis, const float* __restrict__ H,
                           float* __restrict__ O, int e) {
    int t = blockIdx.x * blockDim.x + threadIdx.x;   // E*10 work items (10 float4 per row)
    if (t >= e * 10) return;
    int eg = t / 10;
    int c  = t - eg * 10;
    int s  = src[eg], d = dst[eg];
    float w = dis[s] * dis[d];
    const float4 v = ((const float4*)(H + (size_t)s * OUT_F))[c];
    float* out = O + (size_t)d * OUT_F + c * 4;
    unsafeAtomicAdd(out + 0, v.x * w);
    unsafeAtomicAdd(out + 1, v.y * w);
    unsafeAtomicAdd(out + 2, v.z * w);
    unsafeAtomicAdd(out + 3, v.w * w);
}

// ---------------- host launcher ----------------

extern "C" void kernel_launch(void* const* d_in, const int* in_sizes, int n_in,
                              void* d_out, int out_size, void* d_ws, size_t ws_size,
                              hipStream_t stream) {
    const float* x  = (const float*)d_in[0];
    const float* W1 = (const float*)d_in[1];
    const float* b1 = (const float*)d_in[2];
    const float* W2 = (const float*)d_in[3];
    const float* b2 = (const float*)d_in[4];
    const int* ei   = (const int*)d_in[5];

    const int N = in_sizes[0] / IN_F;
    const int E = in_sizes[5] / 2;
    const int* src = ei;
    const int* dst = ei + E;

    float* out = (float*)d_out;
    float* ws  = (float*)d_ws;

    // workspace layout (floats): dis[N] | H1[N*128] | A1[N*128]; H2 reuses H1
    size_t disOff = 0;
    size_t h1Off  = ((size_t)N + 255) & ~(size_t)255;
    size_t a1Off  = h1Off + (size_t)N * HID_F;
    float* dis = ws + disOff;
    float* H1  = ws + h1Off;
    float* A1  = ws + a1Off;
    float* H2  = H1;   // H1 dead after scatter1

    const int B = 256;
    const int mTiles = (N + 15) / 16;

    // degrees -> dis = deg^{-1/2}
    k_fill1   <<<(N + B - 1) / B, B, 0, stream>>>(dis, N);
    k_degcount<<<(E + B - 1) / B, B, 0, stream>>>(dst, dis, E);
    k_rsqrt   <<<(N + B - 1) / B, B, 0, stream>>>(dis, N);

    // layer 1
    k_gemm1<<<mTiles, 256, 0, stream>>>(x, W1, H1, N);
    {
        int total = N * HID_F;
        k_init1<<<(total + B - 1) / B, B, 0, stream>>>(H1, dis, b1, A1, N);
    }
    {
        int total = E * 32;
        k_scatter1<<<(total + B - 1) / B, B, 0, stream>>>(src, dst, dis, H1, A1, E);
    }

    // layer 2 (relu folded into GEMM2 A-load)
    k_gemm2<<<mTiles, 96, 0, stream>>>(A1, W2, H2, N);
    {
        int total = N * OUT_F;
        k_init2<<<(total + B - 1) / B, B, 0, stream>>>(H2, dis, b2, out, N);
    }
    {
        int total = E * 10;
        k_scatter2<<<(total + B - 1) / B, B, 0, stream>>>(src, dst, dis, H2, out, E);
    }
}